// QLSTMQuantum_65481071399783
// MI455X (gfx1250) — compile-verified
//
#include <hip/hip_runtime.h>
#include <math.h>

typedef __attribute__((ext_vector_type(2))) float v2f;
typedef __attribute__((ext_vector_type(4))) float v4f;
typedef __attribute__((ext_vector_type(8))) float v8f;

#define NQ       8
#define IN_DIM   32
#define HID      8
#define BATCH    1024
#define SEQ      64
#define KDIM     40      // IN_DIM + HID
#define NDIM     32      // 4 gates * 8 qubits
#define GO_STR   20      // go_lds row stride (floats): conflict-free banks, 80B rows (16B aligned)

__device__ __forceinline__ float sigm_f(float x) { return 1.0f / (1.0f + __expf(-x)); }

__global__ __launch_bounds__(32)
void qlstm_wmma_kernel(const float* __restrict__ inputs,
                       const float* __restrict__ Wf, const float* __restrict__ bf, const float* __restrict__ thf,
                       const float* __restrict__ Wi, const float* __restrict__ bi, const float* __restrict__ thi,
                       const float* __restrict__ Wu, const float* __restrict__ bu, const float* __restrict__ thu,
                       const float* __restrict__ Wo, const float* __restrict__ bo, const float* __restrict__ tho,
                       float* __restrict__ out)
{
    __shared__ float h_lds[16][HID];       // hidden state per batch row of this tile
    __shared__ float ytile[16][36];        // 16x32 linear outputs, padded stride 36 (16B-aligned rows)
    __shared__ float go_lds[16][GO_STR];   // g (0..7) and o (8..15) handed from upper to lower lanes

    const int lane = threadIdx.x;          // 0..31 (wave32)
    const int half = lane >> 4;            // 0: gates {f,i}, 1: gates {g,o} in the tail phase
    const int lmod = lane & 15;            // batch row within tile / output column
    const int b0   = blockIdx.x * 16;      // batch tile base

    const float* Wptr[4] = { Wf, Wi, Wu, Wo };
    const float* bptr[4] = { bf, bi, bu, bo };
    const float* tptr[4] = { thf, thi, thu, tho };

    // This lane's two output columns (N0 in tile0, N1 in tile1). Fold b+theta into WMMA C init.
    const int N0 = lmod, N1 = 16 + lmod;
    const float bias0 = bptr[N0 >> 3][N0 & 7] + tptr[N0 >> 3][N0 & 7];
    const float bias1 = bptr[N1 >> 3][N1 & 7] + tptr[N1 >> 3][N1 & 7];

    // Preload B fragments: B[K][N] = W_combined[N][K], K split per WMMA f32 16x16x4 layout.
    // Lane holds B[K = 4*kk + 2*half + {0,1}][N = l%16 (+16 for tile1)].
    v2f Bfrag0[10], Bfrag1[10];
    {
        const float* w0 = Wptr[N0 >> 3] + (size_t)(N0 & 7) * KDIM;
        const float* w1 = Wptr[N1 >> 3] + (size_t)(N1 & 7) * KDIM;
#pragma unroll
        for (int kk = 0; kk < 10; ++kk) {
            const int k0 = 4 * kk + 2 * half;
            Bfrag0[kk] = *(const v2f*)(w0 + k0);
            Bfrag1[kk] = *(const v2f*)(w1 + k0);
        }
    }

    // Tail-phase per-lane activation constants: gp0 -> sigmoid(z); gp1 gateA -> tanh(z)=2*sigm(2z)-1.
    const float mA = half ? 2.0f : 1.0f;
    const float sA = half ? 2.0f : 1.0f;
    const float dA = half ? -1.0f : 0.0f;

    // Init recurrent state.
    if (lane < 16) {
#pragma unroll
        for (int j = 0; j < HID; ++j) h_lds[lane][j] = 0.0f;
    }
    float cx[HID], hx[HID];
#pragma unroll
    for (int j = 0; j < HID; ++j) { cx[j] = 0.0f; hx[j] = 0.0f; }
    __syncthreads();

    // Per-lane global row pointers (lane -> batch row lmod of the tile; +2*half K offset inside row).
    const float* xrow = inputs + (size_t)(b0 + lmod) * (SEQ * IN_DIM) + 2 * half;
    float*       orow = out    + (size_t)(b0 + lmod) * (SEQ * HID);

    for (int t = 0; t < SEQ; ++t) {
        // C init = broadcast (b + theta) down this lane's column.
        v8f acc0, acc1;
#pragma unroll
        for (int v = 0; v < 8; ++v) { acc0[v] = bias0; acc1[v] = bias1; }

        const float* xt = xrow + t * IN_DIM;
        if (t + 1 < SEQ) __builtin_prefetch(xt + IN_DIM, 0, 3);   // global_prefetch_b8 next-step x

        // K-chunks 0..7: x part of comb straight from global (one 128B line per (b,t)).
#pragma unroll
        for (int kk = 0; kk < 8; ++kk) {
            v2f a = *(const v2f*)(xt + 4 * kk);
            acc0 = __builtin_amdgcn_wmma_f32_16x16x4_f32(false, a, false, Bfrag0[kk], (short)0, acc0, false, false);
            acc1 = __builtin_amdgcn_wmma_f32_16x16x4_f32(false, a, false, Bfrag1[kk], (short)0, acc1, false, false);
        }
        // K-chunks 8..9: hidden state from LDS.
#pragma unroll
        for (int kk = 8; kk < 10; ++kk) {
            v2f a = *(const v2f*)(&h_lds[lmod][4 * (kk - 8) + 2 * half]);
            acc0 = __builtin_amdgcn_wmma_f32_16x16x4_f32(false, a, false, Bfrag0[kk], (short)0, acc0, false, false);
            acc1 = __builtin_amdgcn_wmma_f32_16x16x4_f32(false, a, false, Bfrag1[kk], (short)0, acc1, false, false);
        }

        // Spill y tile to LDS: lane owns column N, rows M = 8*half + v.
#pragma unroll
        for (int v = 0; v < 8; ++v) {
            ytile[8 * half + v][N0] = acc0[v];
            ytile[8 * half + v][N1] = acc1[v];
        }
        __syncthreads();

        // ---- Tail stage 1: ALL 32 lanes. Lane pair (r, r+16) shares batch row r = l%16.
        //      half==0 -> gates 0,1 (forget,input); half==1 -> gates 2,3 (update,output).
        const int r = lmod;
        const int col0 = 16 * half;
        float y[16];
        *(v4f*)(y + 0)  = *(const v4f*)(&ytile[r][col0 + 0]);
        *(v4f*)(y + 4)  = *(const v4f*)(&ytile[r][col0 + 4]);
        *(v4f*)(y + 8)  = *(const v4f*)(&ytile[r][col0 + 8]);
        *(v4f*)(y + 12) = *(const v4f*)(&ytile[r][col0 + 12]);

        float vA[NQ], vB[NQ];
#pragma unroll
        for (int g = 0; g < 2; ++g) {
            float c[NQ];
#pragma unroll
            for (int w = 0; w < NQ; ++w) c[w] = __cosf(y[8 * g + w]);
            // E[Z_0] = c1*...*c7 ; E[Z_k] = c0*...*ck (k>=1)  (CNOT chain+ring, Heisenberg picture)
            float z[NQ];
            float p = c[1];
#pragma unroll
            for (int w = 2; w < NQ; ++w) p *= c[w];
            z[0] = p;
            float q = c[0];
#pragma unroll
            for (int w = 1; w < NQ; ++w) { q *= c[w]; z[w] = q; }
            if (g == 0) {
                // gate A: sigmoid on half 0 (f), tanh on half 1 (g) — branch-free affine form
#pragma unroll
                for (int w = 0; w < NQ; ++w) vA[w] = sA * sigm_f(mA * z[w]) + dA;
            } else {
                // gate B: sigmoid for both halves (i / o)
#pragma unroll
                for (int w = 0; w < NQ; ++w) vB[w] = sigm_f(z[w]);
            }
        }
        // Upper lanes hand g (vA) and o (vB) to the lower lanes.
        if (half) {
#pragma unroll
            for (int w = 0; w < NQ; ++w) { go_lds[r][w] = vA[w]; go_lds[r][8 + w] = vB[w]; }
        }
        __syncthreads();

        // ---- Tail stage 2: lower 16 lanes finish the cell (f=vA, i=vB live here).
        if (lane < 16) {
            float gv[NQ], ov[NQ];
            *(v4f*)(gv + 0) = *(const v4f*)(&go_lds[r][0]);
            *(v4f*)(gv + 4) = *(const v4f*)(&go_lds[r][4]);
            *(v4f*)(ov + 0) = *(const v4f*)(&go_lds[r][8]);
            *(v4f*)(ov + 4) = *(const v4f*)(&go_lds[r][12]);
#pragma unroll
            for (int w = 0; w < NQ; ++w) {
                const float c2 = vA[w] * cx[w] + vB[w] * gv[w];
                cx[w] = c2;
                const float th = 2.0f * sigm_f(2.0f * c2) - 1.0f;   // tanh(c2)
                hx[w] = ov[w] * th;
            }
#pragma unroll
            for (int w = 0; w < NQ; ++w) h_lds[r][w] = hx[w];
            // out_seq[b][t][:] — 32B-aligned, two b128 stores
            float* od = orow + t * HID;
            *(v4f*)(od)     = (v4f){ hx[0], hx[1], hx[2], hx[3] };
            *(v4f*)(od + 4) = (v4f){ hx[4], hx[5], hx[6], hx[7] };
        }
        __syncthreads();
    }

    // Final hx / cx tails of d_out.
    if (lane < 16) {
        const int gb = b0 + lane;
        float* ph = out + (size_t)BATCH * SEQ * HID + (size_t)gb * HID;
        float* pc = ph + (size_t)BATCH * HID;
        *(v4f*)(ph)     = (v4f){ hx[0], hx[1], hx[2], hx[3] };
        *(v4f*)(ph + 4) = (v4f){ hx[4], hx[5], hx[6], hx[7] };
        *(v4f*)(pc)     = (v4f){ cx[0], cx[1], cx[2], cx[3] };
        *(v4f*)(pc + 4) = (v4f){ cx[4], cx[5], cx[6], cx[7] };
    }
}

extern "C" void kernel_launch(void* const* d_in, const int* in_sizes, int n_in,
                              void* d_out, int out_size, void* d_ws, size_t ws_size,
                              hipStream_t stream) {
    (void)in_sizes; (void)n_in; (void)out_size; (void)d_ws; (void)ws_size;
    const float* inputs = (const float*)d_in[0];
    const float* Wf  = (const float*)d_in[1];
    const float* bf  = (const float*)d_in[2];
    const float* thf = (const float*)d_in[3];
    const float* Wi  = (const float*)d_in[4];
    const float* bi  = (const float*)d_in[5];
    const float* thi = (const float*)d_in[6];
    const float* Wu  = (const float*)d_in[7];
    const float* bu  = (const float*)d_in[8];
    const float* thu = (const float*)d_in[9];
    const float* Wo  = (const float*)d_in[10];
    const float* bo  = (const float*)d_in[11];
    const float* tho = (const float*)d_in[12];
    float* out = (float*)d_out;

    qlstm_wmma_kernel<<<BATCH / 16, 32, 0, stream>>>(
        inputs, Wf, bf, thf, Wi, bi, thi, Wu, bu, thu, Wo, bo, tho, out);
}